// RegionalTemporalGCN_67242007986550
// MI455X (gfx1250) — compile-verified
//
#include <hip/hip_runtime.h>

typedef __attribute__((ext_vector_type(2))) float v2f;
typedef __attribute__((ext_vector_type(8))) float v8f;

#define N_NODES 10000
#define F_DIM   4
#define P_DIM   12
#define OUT_DIM 256
#define HID_DIM 128
#define O_DIM   12
#define E_MAINN 160000
#define E_REGN  40000

// ---------------- small prep kernels ----------------

__global__ void softmax12_kernel(const float* __restrict__ a, float* __restrict__ probs) {
  if (threadIdx.x == 0) {
    float m = a[0];
    for (int i = 1; i < P_DIM; ++i) m = fmaxf(m, a[i]);
    float e[P_DIM]; float s = 0.0f;
    for (int i = 0; i < P_DIM; ++i) { e[i] = __expf(a[i] - m); s += e[i]; }
    float inv = 1.0f / s;
    for (int i = 0; i < P_DIM; ++i) probs[i] = e[i] * inv;
  }
}

__global__ void zero_f32_kernel(float* __restrict__ p, int count) {
  int i = blockIdx.x * blockDim.x + threadIdx.x;
  if (i < count) p[i] = 0.0f;
}

__global__ void deg_main_kernel(const int* __restrict__ ei, float* __restrict__ deg) {
  int e = blockIdx.x * blockDim.x + threadIdx.x;
  if (e >= E_MAINN) return;
  atomicAdd(&deg[ei[E_MAINN + e]], 1.0f);
}

__global__ void norm_main_kernel(const int* __restrict__ ei, const float* __restrict__ deg,
                                 float* __restrict__ norm) {
  int e = blockIdx.x * blockDim.x + threadIdx.x;
  if (e >= E_MAINN) return;
  int s = ei[e], d = ei[E_MAINN + e];
  norm[e] = rsqrtf(deg[s] + 1.0f) * rsqrtf(deg[d] + 1.0f);
}

__global__ void diag_main_kernel(const float* __restrict__ deg, float* __restrict__ diag) {
  int n = blockIdx.x * blockDim.x + threadIdx.x;
  if (n < N_NODES) diag[n] = 1.0f / (deg[n] + 1.0f);
}

__global__ void deg_reg_kernel(const int* __restrict__ ri, const float* __restrict__ ew,
                               float* __restrict__ deg) {
  int e = blockIdx.x * blockDim.x + threadIdx.x;
  if (e >= E_REGN) return;
  atomicAdd(&deg[ri[e]], ew[e]);
}

__device__ __forceinline__ float dinv_reg(float v) {
  return v > 0.0f ? rsqrtf(fmaxf(v, 1e-12f)) : 0.0f;
}

__global__ void norm_reg_kernel(const int* __restrict__ ri, const float* __restrict__ ew,
                                const float* __restrict__ deg, float* __restrict__ norm) {
  int e = blockIdx.x * blockDim.x + threadIdx.x;
  if (e >= E_REGN) return;
  norm[e] = dinv_reg(deg[ri[e]]) * ew[e] * dinv_reg(deg[ri[E_REGN + e]]);
}

// ---------------- one-time weight folds / packing ----------------

// M0 = W0 @ (sum_r Wcat_r)   (4x256)
// Mr[r] = W1 @ Wcat_r        (5 x 4x256)
// biash = cheb_b @ (sum_r Wcat_r) + lin_cat_b
__global__ void combine_cheb_weights_kernel(const float* __restrict__ W0, const float* __restrict__ W1,
                                            const float* __restrict__ chebb,
                                            const float* __restrict__ LC, const float* __restrict__ lcb,
                                            float* __restrict__ M0, float* __restrict__ Mr,
                                            float* __restrict__ biash) {
  int j = threadIdx.x;           // 256 threads, 1 block
  float m0[4] = {0.f, 0.f, 0.f, 0.f};
  float mr[5][4] = {};
  float bh = lcb[j];
  for (int k = 0; k < OUT_DIM; ++k) {
    float s = 0.0f;
    float cb = chebb[k];
#pragma unroll
    for (int r = 0; r < 5; ++r) {
      float lw = LC[(r * OUT_DIM + k) * OUT_DIM + j];
      s += lw;
#pragma unroll
      for (int f = 0; f < 4; ++f) mr[r][f] += W1[f * OUT_DIM + k] * lw;
    }
#pragma unroll
    for (int f = 0; f < 4; ++f) m0[f] += W0[f * OUT_DIM + k] * s;
    bh += cb * s;
  }
#pragma unroll
  for (int f = 0; f < 4; ++f) M0[f * OUT_DIM + j] = m0[f];
#pragma unroll
  for (int r = 0; r < 5; ++r)
#pragma unroll
    for (int f = 0; f < 4; ++f) Mr[(r * 4 + f) * OUT_DIM + j] = mr[r][f];
  biash[j] = bh;
}

// Gp (pair-packed 4x256) = gW @ lW[:256],  bout = gb @ lW[:256] + lb
// packed layout: Gp[((f>>1)*256 + j)*2 + (f&1)]
__global__ void combine_gate_weights_kernel(const float* __restrict__ gW, const float* __restrict__ gb,
                                            const float* __restrict__ lW, const float* __restrict__ lb,
                                            float* __restrict__ Gp, float* __restrict__ bout) {
  int j = threadIdx.x;           // 256 threads, 1 block
  float g[4] = {0.f, 0.f, 0.f, 0.f};
  float b = lb[j];
  for (int k = 0; k < OUT_DIM; ++k) {
    float lw = lW[k * OUT_DIM + j];
#pragma unroll
    for (int f = 0; f < 4; ++f) g[f] += gW[f * OUT_DIM + k] * lw;
    b += gb[k] * lw;
  }
#pragma unroll
  for (int f = 0; f < 4; ++f) Gp[((f >> 1) * OUT_DIM + j) * 2 + (f & 1)] = g[f];
  bout[j] = b;
}

// pair-pack a KxNC row-major weight: Bp[((k>>1)*NC + j)*2 + (k&1)] = B[k*NC + j]
__global__ void pack_pairs_kernel(const float* __restrict__ B, float* __restrict__ Bp,
                                  int K, int NC) {
  int i = blockIdx.x * blockDim.x + threadIdx.x;
  if (i >= K * NC) return;
  int k = i / NC, j = i - k * NC;
  Bp[((k >> 1) * NC + j) * 2 + (k & 1)] = B[i];
}

// ---------------- per-step sparse kernels ----------------

// Yp = diag * Xp ; Tx1 = 0
__global__ void init_p_kernel(const float* __restrict__ x, int p, const float* __restrict__ diag,
                              float* __restrict__ Yp, float* __restrict__ Tx1) {
  int i = blockIdx.x * blockDim.x + threadIdx.x;
  if (i < N_NODES * 4) {
    int n = i >> 2, f = i & 3;
    Yp[i] = diag[n] * x[(n * F_DIM + f) * P_DIM + p];
  }
  if (i < 5 * N_NODES * 4) Tx1[i] = 0.0f;
}

__global__ void scatter_main_kernel(const int* __restrict__ ei, const float* __restrict__ norm,
                                    const float* __restrict__ x, int p, float* __restrict__ Yp) {
  int e = blockIdx.x * blockDim.x + threadIdx.x;
  if (e >= E_MAINN) return;
  int s = ei[e], d = ei[E_MAINN + e];
  float w = norm[e];
#pragma unroll
  for (int f = 0; f < 4; ++f)
    atomicAdd(&Yp[d * 4 + f], w * x[(s * F_DIM + f) * P_DIM + p]);
}

__global__ void scatter_reg_kernel(const int* __restrict__ ri, const float* __restrict__ norm,
                                   const float* __restrict__ x, int p, float* __restrict__ Tr) {
  int e = blockIdx.x * blockDim.x + threadIdx.x;
  if (e >= E_REGN) return;
  int s = ri[e], d = ri[E_REGN + e];
  float w = -norm[e];   // Tx1 = -segment_sum(...)
#pragma unroll
  for (int f = 0; f < 4; ++f)
    atomicAdd(&Tr[d * 4 + f], w * x[(s * F_DIM + f) * P_DIM + p]);
}

// h = leaky_relu(Xp@M0 + sum_r Tx1_r@Mr + biash)
__global__ void cheb_cat_kernel(const float* __restrict__ x, int p, const float* __restrict__ Tx1,
                                const float* __restrict__ M0, const float* __restrict__ Mr,
                                const float* __restrict__ biash, float* __restrict__ h) {
  __shared__ float sX[4];
  __shared__ float sT[20];
  int n = blockIdx.x;
  int j = threadIdx.x;
  if (j < 4) sX[j] = x[(n * F_DIM + j) * P_DIM + p];
  else if (j < 24) {
    int t = j - 4;
    sT[t] = Tx1[((t >> 2) * N_NODES + n) * 4 + (t & 3)];
  }
  __syncthreads();
  float acc = biash[j];
#pragma unroll
  for (int f = 0; f < 4; ++f) acc += sX[f] * M0[f * OUT_DIM + j];
#pragma unroll
  for (int r = 0; r < 5; ++r)
#pragma unroll
    for (int f = 0; f < 4; ++f)
      acc += sT[r * 4 + f] * Mr[(r * 4 + f) * OUT_DIM + j];
  h[n * OUT_DIM + j] = (acc > 0.0f) ? acc : 0.01f * acc;
}

// ---------------- fp32 WMMA GEMM ----------------
// Out = act( A(Nx256) @ B(256xNC) + Y(Nx4) @ G(4xNC) + bias )
// B and G are pair-packed:  Bp[((k>>1)*NC + j)*2 + (k&1)]
// Wave computes 16 rows x (CT*16) cols, CT = NC/128.  Block = 8 waves = all NC cols.
// MODE: 0=sigmoid  1=sigmoid * Hm  2=tanh  3=relu
template <int MODE, int NC>
__global__ void wmma_gemm_kernel(const float* __restrict__ A, const float* __restrict__ Bp,
                                 const float* __restrict__ Gp, const float* __restrict__ Y,
                                 const float* __restrict__ bias, const float* __restrict__ Hm,
                                 float* __restrict__ Out) {
  constexpr int CT = NC / 128;         // column tiles per wave (2 for 256, 1 for 128)
  const int lane = threadIdx.x & 31;
  const int wave = threadIdx.x >> 5;
  const int row  = lane & 15;          // A row within tile / B col within tile
  const int kd   = (lane >> 4) << 1;   // 0 for lanes 0-15, 2 for lanes 16-31
  const int m0   = blockIdx.x << 4;    // 16-row tile base
  const int col0 = wave * (16 * CT) + row;

  v8f c[CT];
#pragma unroll
  for (int t = 0; t < CT; ++t) {
    float bj = bias[col0 + 16 * t];
#pragma unroll
    for (int i = 0; i < 8; ++i) c[t][i] = bj;
  }

  if (Gp != nullptr) {                 // fold the K=4 Yp term in as one WMMA per tile
    v2f a = *(const v2f*)(Y + (m0 + row) * 4 + kd);
#pragma unroll
    for (int t = 0; t < CT; ++t) {
      v2f b = *(const v2f*)(Gp + kd * NC + (col0 + 16 * t) * 2);
      c[t] = __builtin_amdgcn_wmma_f32_16x16x4_f32(false, a, false, b, (short)0, c[t], false, false);
    }
  }

  const float* Abase = A + (m0 + row) * OUT_DIM + kd;
  const float* Bbase = Bp + kd * NC + col0 * 2;   // + k0*NC per k-step (immediate offsets)
#pragma unroll
  for (int k0 = 0; k0 < OUT_DIM; k0 += 4) {
    v2f a = *(const v2f*)(Abase + k0);
#pragma unroll
    for (int t = 0; t < CT; ++t) {
      v2f b = *(const v2f*)(Bbase + k0 * NC + t * 32);
      c[t] = __builtin_amdgcn_wmma_f32_16x16x4_f32(false, a, false, b, (short)0, c[t], false, false);
    }
  }

#pragma unroll
  for (int t = 0; t < CT; ++t) {
    int col = col0 + 16 * t;
#pragma unroll
    for (int i = 0; i < 8; ++i) {
      int r = m0 + ((lane < 16) ? i : (i + 8));
      float v = c[t][i];
      if (MODE == 0)      v = 1.0f / (1.0f + __expf(-v));
      else if (MODE == 1) v = Hm[r * OUT_DIM + col] * (1.0f / (1.0f + __expf(-v)));
      else if (MODE == 2) v = tanhf(v);
      else                v = fmaxf(v, 0.0f);
      Out[r * NC + col] = v;
    }
  }
}

// H += probs[p] * (Z*h + (1-Z)*T)
__global__ void gru_update_kernel(const float* __restrict__ h, const float* __restrict__ Z,
                                  const float* __restrict__ T, const float* __restrict__ probs,
                                  int p, float* __restrict__ H) {
  int i = blockIdx.x * blockDim.x + threadIdx.x;
  if (i >= N_NODES * OUT_DIM) return;
  float z = Z[i];
  H[i] += probs[p] * (z * h[i] + (1.0f - z) * T[i]);
}

__global__ void relu_kernel(const float* __restrict__ in, float* __restrict__ out, int count) {
  int i = blockIdx.x * blockDim.x + threadIdx.x;
  if (i < count) out[i] = fmaxf(in[i], 0.0f);
}

// o = t(Nx128) @ W2(128x12) + b2
__global__ void out2_kernel(const float* __restrict__ t, const float* __restrict__ W2,
                            const float* __restrict__ b2, float* __restrict__ o) {
  int i = blockIdx.x * blockDim.x + threadIdx.x;
  if (i >= N_NODES * O_DIM) return;
  int n = i / O_DIM, d = i - n * O_DIM;
  float acc = b2[d];
  for (int k = 0; k < HID_DIM; ++k) acc += t[n * HID_DIM + k] * W2[k * O_DIM + d];
  o[i] = acc;
}

// ---------------- launch ----------------

extern "C" void kernel_launch(void* const* d_in, const int* in_sizes, int n_in,
                              void* d_out, int out_size, void* d_ws, size_t ws_size,
                              hipStream_t stream) {
  (void)in_sizes; (void)n_in; (void)out_size; (void)ws_size;

  const float* x  = (const float*)d_in[0];
  const int*   ei = (const int*)d_in[1];
  const int*   ri[5]; const float* rw[5];
  for (int r = 0; r < 5; ++r) { ri[r] = (const int*)d_in[2 + 2 * r]; rw[r] = (const float*)d_in[3 + 2 * r]; }
  const float* cheb_W0   = (const float*)d_in[12];
  const float* cheb_W1   = (const float*)d_in[13];
  const float* cheb_b    = (const float*)d_in[14];
  const float* lin_cat_W = (const float*)d_in[15];
  const float* lin_cat_b = (const float*)d_in[16];
  const float* attention = (const float*)d_in[17];
  const float* gW[3] = {(const float*)d_in[18], (const float*)d_in[20], (const float*)d_in[22]};
  const float* gb[3] = {(const float*)d_in[19], (const float*)d_in[21], (const float*)d_in[23]};
  const float* lW[3] = {(const float*)d_in[24], (const float*)d_in[26], (const float*)d_in[28]};
  const float* lb[3] = {(const float*)d_in[25], (const float*)d_in[27], (const float*)d_in[29]};
  const float* out1_W = (const float*)d_in[30];
  const float* out1_b = (const float*)d_in[31];
  const float* out2_W = (const float*)d_in[32];
  const float* out2_b = (const float*)d_in[33];

  float* ws = (float*)d_ws;
  size_t off = 0;
  auto alloc = [&](size_t n) { float* p = ws + off; off += (n + 63) & ~(size_t)63; return p; };
  float* probs     = alloc(16);
  float* deg_main  = alloc(N_NODES);
  float* deg_reg   = alloc(5 * N_NODES);
  float* norm_main = alloc(E_MAINN);
  float* diag_main = alloc(N_NODES);
  float* norm_reg  = alloc(5 * E_REGN);
  float* M0        = alloc(4 * OUT_DIM);
  float* Mr        = alloc(5 * 4 * OUT_DIM);
  float* biash     = alloc(OUT_DIM);
  float* Gp3       = alloc(3 * 4 * OUT_DIM);                   // pair-packed gate folds
  float* b3        = alloc(3 * OUT_DIM);
  float* Bp3       = alloc((size_t)3 * OUT_DIM * OUT_DIM);     // pair-packed Lz1/Lr1/Lh1
  float* out1p     = alloc((size_t)OUT_DIM * HID_DIM);         // pair-packed out1_W
  float* Tx1       = alloc(5 * N_NODES * 4);
  float* Yp        = alloc(N_NODES * 4);
  float* hbuf      = alloc((size_t)N_NODES * OUT_DIM);
  float* Zbuf      = alloc((size_t)N_NODES * OUT_DIM);
  float* Rbuf      = alloc((size_t)N_NODES * OUT_DIM);
  float* Tbuf      = alloc((size_t)N_NODES * OUT_DIM);
  float* hr   = Zbuf;   // reuse after p-loop
  float* thid = Rbuf;   // reuse after p-loop

  float* o = (float*)d_out;                       // N x 12
  float* H = (float*)d_out + N_NODES * O_DIM;     // N x 256 (out_hidden, accumulated in place)

  const int TB = 256;
  auto nb = [](int n) { return (n + 255) / 256; };

  // ---- one-time prep ----
  softmax12_kernel<<<1, 32, 0, stream>>>(attention, probs);
  zero_f32_kernel<<<nb(N_NODES), TB, 0, stream>>>(deg_main, N_NODES);
  zero_f32_kernel<<<nb(5 * N_NODES), TB, 0, stream>>>(deg_reg, 5 * N_NODES);
  deg_main_kernel<<<nb(E_MAINN), TB, 0, stream>>>(ei, deg_main);
  for (int r = 0; r < 5; ++r)
    deg_reg_kernel<<<nb(E_REGN), TB, 0, stream>>>(ri[r], rw[r], deg_reg + r * N_NODES);
  norm_main_kernel<<<nb(E_MAINN), TB, 0, stream>>>(ei, deg_main, norm_main);
  diag_main_kernel<<<nb(N_NODES), TB, 0, stream>>>(deg_main, diag_main);
  for (int r = 0; r < 5; ++r)
    norm_reg_kernel<<<nb(E_REGN), TB, 0, stream>>>(ri[r], rw[r], deg_reg + r * N_NODES,
                                                   norm_reg + r * E_REGN);
  combine_cheb_weights_kernel<<<1, 256, 0, stream>>>(cheb_W0, cheb_W1, cheb_b, lin_cat_W, lin_cat_b,
                                                     M0, Mr, biash);
  for (int g = 0; g < 3; ++g) {
    combine_gate_weights_kernel<<<1, 256, 0, stream>>>(gW[g], gb[g], lW[g], lb[g],
                                                       Gp3 + g * 1024, b3 + g * 256);
    pack_pairs_kernel<<<nb(OUT_DIM * OUT_DIM), TB, 0, stream>>>(lW[g] + OUT_DIM * OUT_DIM,
                                                                Bp3 + (size_t)g * OUT_DIM * OUT_DIM,
                                                                OUT_DIM, OUT_DIM);
  }
  pack_pairs_kernel<<<nb(OUT_DIM * HID_DIM), TB, 0, stream>>>(out1_W, out1p, OUT_DIM, HID_DIM);
  zero_f32_kernel<<<nb(N_NODES * OUT_DIM), TB, 0, stream>>>(H, N_NODES * OUT_DIM);

  // ---- per-timestep loop ----
  dim3 ggrid(N_NODES / 16);            // 625 row-tiles, one block covers all 256 cols
  for (int p = 0; p < P_DIM; ++p) {
    init_p_kernel<<<nb(5 * N_NODES * 4), TB, 0, stream>>>(x, p, diag_main, Yp, Tx1);
    scatter_main_kernel<<<nb(E_MAINN), TB, 0, stream>>>(ei, norm_main, x, p, Yp);
    for (int r = 0; r < 5; ++r)
      scatter_reg_kernel<<<nb(E_REGN), TB, 0, stream>>>(ri[r], norm_reg + r * E_REGN, x, p,
                                                        Tx1 + (size_t)r * N_NODES * 4);
    cheb_cat_kernel<<<N_NODES, 256, 0, stream>>>(x, p, Tx1, M0, Mr, biash, hbuf);
    // Z = sigmoid(h@Lz1 + Yp@Gz + bz)
    wmma_gemm_kernel<0, 256><<<ggrid, 256, 0, stream>>>(hbuf, Bp3, Gp3, Yp,
                                                        b3, nullptr, Zbuf);
    // Rbuf = h * sigmoid(h@Lr1 + Yp@Gr + br)
    wmma_gemm_kernel<1, 256><<<ggrid, 256, 0, stream>>>(hbuf, Bp3 + (size_t)OUT_DIM * OUT_DIM,
                                                        Gp3 + 1024, Yp, b3 + 256, hbuf, Rbuf);
    // T = tanh((h*R)@Lh1 + Yp@Gh + bh)
    wmma_gemm_kernel<2, 256><<<ggrid, 256, 0, stream>>>(Rbuf, Bp3 + (size_t)2 * OUT_DIM * OUT_DIM,
                                                        Gp3 + 2048, Yp, b3 + 512, nullptr, Tbuf);
    gru_update_kernel<<<nb(N_NODES * OUT_DIM), TB, 0, stream>>>(hbuf, Zbuf, Tbuf, probs, p, H);
  }

  // ---- output head ----
  relu_kernel<<<nb(N_NODES * OUT_DIM), TB, 0, stream>>>(H, hr, N_NODES * OUT_DIM);
  wmma_gemm_kernel<3, 128><<<ggrid, 256, 0, stream>>>(hr, out1p, nullptr, nullptr, out1_b,
                                                      nullptr, thid);
  out2_kernel<<<nb(N_NODES * O_DIM), TB, 0, stream>>>(thid, out2_W, out2_b, o);
}